// DNC_4346506904349
// MI455X (gfx1250) — compile-verified
//
#include <hip/hip_runtime.h>
#include <math.h>

#define DEV __device__ __forceinline__

typedef __attribute__((ext_vector_type(2))) float v2f;
typedef __attribute__((ext_vector_type(8))) float v8f;

// ---- constants (match reference) ----
// B=64 T=32 IN=256 H=512 OUT=256 M=256 W=64 R=4 NN_IN=512 NN_OUT=768 IFACE=471
#define DELTA_ 1e-6f

DEV v8f zero8() { v8f z = {0.f,0.f,0.f,0.f,0.f,0.f,0.f,0.f}; return z; }

DEV v8f wmma4(v2f a, v2f b, v8f c) {
  // D = A(16x4,f32) * B(4x16,f32) + C(16x16,f32), wave32
  return __builtin_amdgcn_wmma_f32_16x16x4_f32(false, a, false, b, (short)0, c, false, false);
}

DEV float sigmoidf_(float x) { return 1.0f / (1.0f + expf(-x)); }
DEV float softplusf_(float x) { return fmaxf(x, 0.f) + log1pf(expf(-fabsf(x))); }
DEV float clip20(float x) { return fminf(fmaxf(x, -20.f), 20.f); }

DEV float blockReduceMax(float v, float* red, int tid) {
  red[tid] = v; __syncthreads();
  for (int s = 128; s > 0; s >>= 1) { if (tid < s) red[tid] = fmaxf(red[tid], red[tid + s]); __syncthreads(); }
  float r = red[0]; __syncthreads(); return r;
}
DEV float blockReduceSum(float v, float* red, int tid) {
  red[tid] = v; __syncthreads();
  for (int s = 128; s > 0; s >>= 1) { if (tid < s) red[tid] += red[tid + s]; __syncthreads(); }
  float r = red[0]; __syncthreads(); return r;
}

// ---------------------------------------------------------------------------
// Fused LSTM step: g = A1@W1^T + Hprev@Whh^T + bih + bhh ; gates; c,h update.
// 128 waves: rt in [0,4) row tiles (64 rows), cg in [0,32) col groups (512 cols),
// each wave holds 4 accumulators (gates i,f,g,o at cols cg*16 + gate*512).
// ---------------------------------------------------------------------------
__global__ __launch_bounds__(256) void lstm_kernel(
    const float* __restrict__ A1, int lda1, int Ka,
    const float* __restrict__ Wih,       // (2048, 512) row-major
    const float* __restrict__ Hprev,     // (64, 512)
    const float* __restrict__ Whh,       // (2048, 512)
    const float* __restrict__ bih, const float* __restrict__ bhh,
    const float* __restrict__ cin, float* __restrict__ cout,
    float* __restrict__ hout)
{
  const int wv   = (blockIdx.x * blockDim.x + threadIdx.x) >> 5;
  const int lane = threadIdx.x & 31;
  const int half = lane >> 4, l16 = lane & 15;
  const int rt = wv >> 5, cg = wv & 31;
  const int arow = rt * 16 + l16;
  const int cb = cg * 16 + l16;
  const int koff = 2 * half;

  v8f acc0 = zero8(), acc1 = zero8(), acc2 = zero8(), acc3 = zero8();

  for (int k0 = 0; k0 < Ka; k0 += 4) {
    v2f a = *(const v2f*)(A1 + (long)arow * lda1 + k0 + koff);
    v2f b0 = *(const v2f*)(Wih + (0    + cb) * 512 + k0 + koff);
    v2f b1 = *(const v2f*)(Wih + (512  + cb) * 512 + k0 + koff);
    v2f b2 = *(const v2f*)(Wih + (1024 + cb) * 512 + k0 + koff);
    v2f b3 = *(const v2f*)(Wih + (1536 + cb) * 512 + k0 + koff);
    acc0 = wmma4(a, b0, acc0); acc1 = wmma4(a, b1, acc1);
    acc2 = wmma4(a, b2, acc2); acc3 = wmma4(a, b3, acc3);
  }
  for (int k0 = 0; k0 < 512; k0 += 4) {
    v2f a = *(const v2f*)(Hprev + arow * 512 + k0 + koff);
    v2f b0 = *(const v2f*)(Whh + (0    + cb) * 512 + k0 + koff);
    v2f b1 = *(const v2f*)(Whh + (512  + cb) * 512 + k0 + koff);
    v2f b2 = *(const v2f*)(Whh + (1024 + cb) * 512 + k0 + koff);
    v2f b3 = *(const v2f*)(Whh + (1536 + cb) * 512 + k0 + koff);
    acc0 = wmma4(a, b0, acc0); acc1 = wmma4(a, b1, acc1);
    acc2 = wmma4(a, b2, acc2); acc3 = wmma4(a, b3, acc3);
  }

  const float bs0 = bih[cb] + bhh[cb];
  const float bs1 = bih[512 + cb] + bhh[512 + cb];
  const float bs2 = bih[1024 + cb] + bhh[1024 + cb];
  const float bs3 = bih[1536 + cb] + bhh[1536 + cb];

#pragma unroll
  for (int v = 0; v < 8; ++v) {
    int r = rt * 16 + v + 8 * half;
    int idx = r * 512 + cb;
    float gi = sigmoidf_(acc0[v] + bs0);
    float gf = sigmoidf_(acc1[v] + bs1);
    float gg = tanhf(acc2[v] + bs2);
    float go = sigmoidf_(acc3[v] + bs3);
    float cN = gf * cin[idx] + gi * gg;
    cout[idx] = cN;
    hout[idx] = go * tanhf(cN);
  }
}

// ---------------------------------------------------------------------------
// xi = clip(h1) @ Wif^T + bif     (64 x 471, padded to 480 cols)
// 120 waves: rt in [0,4), ct in [0,30)
// ---------------------------------------------------------------------------
__global__ __launch_bounds__(256) void iface_kernel(
    const float* __restrict__ h1, const float* __restrict__ Wif,
    const float* __restrict__ bif, float* __restrict__ xi)
{
  const int wv   = (blockIdx.x * blockDim.x + threadIdx.x) >> 5;
  const int lane = threadIdx.x & 31;
  const int half = lane >> 4, l16 = lane & 15;
  const int rt = wv / 30, ct = wv % 30;
  const int arow = rt * 16 + l16;
  const int col = ct * 16 + l16;
  const int colc = (col < 471) ? col : 0;
  const int koff = 2 * half;

  v8f acc = zero8();
  for (int k0 = 0; k0 < 512; k0 += 4) {
    v2f a = *(const v2f*)(h1 + arow * 512 + k0 + koff);
    a[0] = clip20(a[0]); a[1] = clip20(a[1]);
    v2f b = *(const v2f*)(Wif + colc * 512 + k0 + koff);
    acc = wmma4(a, b, acc);
  }
  if (col < 471) {
    float bb = bif[col];
#pragma unroll
    for (int v = 0; v < 8; ++v) {
      int r = rt * 16 + v + 8 * half;
      xi[r * 480 + col] = acc[v] + bb;
    }
  }
}

// ---------------------------------------------------------------------------
// final: out[b,t,:] = tanh(ys[t*64+b,:]) @ Wout^T + bout   (2048 x 256, K=768)
// 2048 waves: rt in [0,128), ct in [0,16)
// ---------------------------------------------------------------------------
__global__ __launch_bounds__(256) void outgemm_kernel(
    const float* __restrict__ ys, const float* __restrict__ Wout,
    const float* __restrict__ bout, float* __restrict__ out)
{
  const int wv   = (blockIdx.x * blockDim.x + threadIdx.x) >> 5;
  const int lane = threadIdx.x & 31;
  const int half = lane >> 4, l16 = lane & 15;
  const int rt = wv >> 4, ct = wv & 15;
  const int arow = rt * 16 + l16;
  const int col = ct * 16 + l16;
  const int koff = 2 * half;

  v8f acc = zero8();
  for (int k0 = 0; k0 < 768; k0 += 4) {
    v2f a = *(const v2f*)(ys + (long)arow * 768 + k0 + koff);
    a[0] = tanhf(a[0]); a[1] = tanhf(a[1]);
    v2f b = *(const v2f*)(Wout + col * 768 + k0 + koff);
    acc = wmma4(a, b, acc);
  }
  float bb = bout[col];
#pragma unroll
  for (int v = 0; v < 8; ++v) {
    int r = rt * 16 + v + 8 * half;   // r = t*64 + b
    int tt = r >> 6, bidx = r & 63;
    out[(bidx * 32 + tt) * 256 + col] = acc[v] + bb;
  }
}

// ---------------------------------------------------------------------------
// M1: head nonlinearities, usage update, write content weights, allocation
//     (stable sort), write weighting, memory erase+write. One block per batch.
// ---------------------------------------------------------------------------
__global__ __launch_bounds__(256) void mem_write_kernel(
    const float* __restrict__ xi, float* __restrict__ mem,
    const float* __restrict__ rw, float* __restrict__ ww,
    float* __restrict__ usage, float* __restrict__ wwsum,
    float* __restrict__ hrk, float* __restrict__ hrs, float* __restrict__ hrm)
{
  const int b = blockIdx.x, tid = threadIdx.x, m = tid;
  __shared__ float wk_s[64], ev_s[64], wv_s[64], fg_s[4], sc_s[3];
  __shared__ float sk[256]; __shared__ int si[256];
  __shared__ float excl[256], wcw_s[256], alloc_s[256], ww_s[256], red[256];

  const float* xb = xi + b * 480;
  hrk[b * 256 + tid] = tanhf(xb[tid]);                               // rk (4x64) tanh
  if (tid < 4) hrs[b * 4 + tid] = softplusf_(xb[256 + tid]);         // rs softplus
  if (tid < 64) {
    wk_s[tid] = tanhf(xb[260 + tid]);                                // wk tanh
    ev_s[tid] = sigmoidf_(xb[325 + tid]);                            // erase vec
    wv_s[tid] = tanhf(xb[389 + tid]);                                // write vec
  }
  if (tid < 4) fg_s[tid] = sigmoidf_(xb[453 + tid]);                 // free gates
  if (tid == 0) {
    sc_s[0] = softplusf_(xb[324]);                                   // write strength
    sc_s[1] = sigmoidf_(xb[457]);                                    // alloc gate
    sc_s[2] = sigmoidf_(xb[458]);                                    // write gate
  }
  if (tid < 4) {                                                     // read modes softmax(3)
    float a0 = xb[459 + 3 * tid], a1 = xb[460 + 3 * tid], a2 = xb[461 + 3 * tid];
    float mx = fmaxf(a0, fmaxf(a1, a2));
    float e0 = expf(a0 - mx), e1 = expf(a1 - mx), e2 = expf(a2 - mx);
    float s = e0 + e1 + e2;
    hrm[b * 12 + 3 * tid] = e0 / s; hrm[b * 12 + 3 * tid + 1] = e1 / s; hrm[b * 12 + 3 * tid + 2] = e2 / s;
  }
  __syncthreads();

  // usage update (OLD ww, OLD rw)
  float u = usage[b * 256 + m];
  float wwo = ww[b * 256 + m];
  u = u + (1.f - u) * wwo;
  float psi = 1.f;
#pragma unroll
  for (int r = 0; r < 4; ++r) psi *= (1.f - fg_s[r] * rw[b * 1024 + r * 256 + m]);
  u *= psi;
  usage[b * 256 + m] = u;

  // write content weights on OLD memory
  float nk = 0.f;
  for (int w2 = 0; w2 < 64; ++w2) nk += wk_s[w2] * wk_s[w2];
  nk = sqrtf(nk);
  float dot = 0.f, nm = 0.f;
  const float* mrow = mem + (size_t)b * 16384 + m * 64;
  for (int w2 = 0; w2 < 64; ++w2) { float mv = mrow[w2]; dot += wk_s[w2] * mv; nm += mv * mv; }
  nm = sqrtf(nm);
  float sim = dot / ((nm + DELTA_) * (nk + DELTA_)) * sc_s[0];
  float mx = blockReduceMax(sim, red, tid);
  float e = expf(sim - mx);
  float ssum = blockReduceSum(e, red, tid);
  wcw_s[m] = e / ssum;

  // allocation: stable ascending sort of u-keys (bitonic, tie-break on index)
  sk[m] = DELTA_ + (1.f - DELTA_) * u; si[m] = m;
  __syncthreads();
  for (int kk = 2; kk <= 256; kk <<= 1) {
    for (int j = kk >> 1; j > 0; j >>= 1) {
      int ixj = tid ^ j;
      if (ixj > tid) {
        float a = sk[tid], bb = sk[ixj]; int ia = si[tid], ib = si[ixj];
        bool asc = ((tid & kk) == 0);
        bool gt = (a > bb) || (a == bb && ia > ib);
        bool sw = asc ? gt : !gt;
        if (sw) { sk[tid] = bb; sk[ixj] = a; si[tid] = ib; si[ixj] = ia; }
      }
      __syncthreads();
    }
  }
  if (tid == 0) { float p = 1.f; for (int j = 0; j < 256; ++j) { excl[j] = p; p *= sk[j]; } }
  __syncthreads();
  alloc_s[si[tid]] = (1.f - sk[tid]) * excl[tid];
  __syncthreads();

  // new write weighting
  float wwn = sc_s[2] * (sc_s[1] * alloc_s[m] + (1.f - sc_s[1]) * wcw_s[m]);
  ww[b * 256 + m] = wwn; ww_s[m] = wwn;
  float wsum = blockReduceSum(wwn, red, tid);
  if (tid == 0) wwsum[b] = wsum;
  __syncthreads();

  // memory erase + write
  float wm = ww_s[m];
  float* mroww = mem + (size_t)b * 16384 + m * 64;
  for (int w2 = 0; w2 < 64; ++w2)
    mroww[w2] = mroww[w2] * (1.f - wm * ev_s[w2]) + wm * wv_s[w2];
}

// link = (1 - ww_i - ww_j)*link + ww_i*prec_j ; zero diagonal. (prec is OLD)
__global__ void link_kernel(float* __restrict__ link, const float* __restrict__ ww,
                            const float* __restrict__ prec)
{
  int idx = blockIdx.x * 256 + threadIdx.x;          // 64*256*256 total
  int b = idx >> 16, ij = idx & 65535, i = ij >> 8, j = ij & 255;
  float wi = ww[b * 256 + i], wj = ww[b * 256 + j], pj = prec[b * 256 + j];
  float L = link[idx];
  L = (1.f - wi - wj) * L + wi * pj;
  link[idx] = (i == j) ? 0.f : L;
}

// prec = (1 - sum(ww))*prec + ww   (runs after link_kernel)
__global__ void prec_kernel(float* __restrict__ prec, const float* __restrict__ ww,
                            const float* __restrict__ wwsum)
{
  int idx = blockIdx.x * 256 + threadIdx.x;
  int b = idx >> 8;
  prec[idx] = (1.f - wwsum[b]) * prec[idx] + ww[idx];
}

// ---------------------------------------------------------------------------
// M3: read content weights (NEW mem), fwd/bwd via NEW link and OLD rw, blend,
//     read vectors, pack o_t = [clip(h1), rvec] into ys. One block per batch.
// ---------------------------------------------------------------------------
__global__ __launch_bounds__(256) void read_kernel(
    const float* __restrict__ mem, const float* __restrict__ link,
    float* __restrict__ rw, const float* __restrict__ hrk,
    const float* __restrict__ hrs, const float* __restrict__ hrm,
    const float* __restrict__ h1, float* __restrict__ ys_t)
{
  const int b = blockIdx.x, tid = threadIdx.x, m = tid;
  __shared__ float rwold_s[1024], rwn_s[1024], norm_s[256], rk_s[64], red[256], rm_s[12], rs_s[4];

#pragma unroll
  for (int q = 0; q < 4; ++q) rwold_s[q * 256 + tid] = rw[b * 1024 + q * 256 + tid];
  if (tid < 12) rm_s[tid] = hrm[b * 12 + tid];
  if (tid < 4)  rs_s[tid] = hrs[b * 4 + tid];
  {
    const float* mrow = mem + (size_t)b * 16384 + m * 64;
    float s2 = 0.f;
    for (int w2 = 0; w2 < 64; ++w2) s2 += mrow[w2] * mrow[w2];
    norm_s[m] = sqrtf(s2);
  }
  __syncthreads();

  // forward / backward weights
  float fwd[4] = {0.f, 0.f, 0.f, 0.f}, bwd[4] = {0.f, 0.f, 0.f, 0.f};
  const float* Lb = link + (size_t)b * 65536;
  for (int j = 0; j < 256; ++j) {
    float lr = Lb[m * 256 + j];       // link[i=m][j]
    float lc = Lb[j * 256 + m];       // link[j][i=m]
#pragma unroll
    for (int r = 0; r < 4; ++r) { float rj = rwold_s[r * 256 + j]; fwd[r] += lr * rj; bwd[r] += rj * lc; }
  }

  for (int r = 0; r < 4; ++r) {
    if (tid < 64) rk_s[tid] = hrk[b * 256 + r * 64 + tid];
    __syncthreads();
    float nk = 0.f;
    for (int w2 = 0; w2 < 64; ++w2) nk += rk_s[w2] * rk_s[w2];
    nk = sqrtf(nk);
    float dot = 0.f;
    const float* mrow = mem + (size_t)b * 16384 + m * 64;
    for (int w2 = 0; w2 < 64; ++w2) dot += rk_s[w2] * mrow[w2];
    float sim = dot / ((norm_s[m] + DELTA_) * (nk + DELTA_)) * rs_s[r];
    float mx = blockReduceMax(sim, red, tid);
    float e = expf(sim - mx);
    float ss = blockReduceSum(e, red, tid);
    float rcw = e / ss;
    float rwn = rm_s[3 * r] * bwd[r] + rm_s[3 * r + 1] * fwd[r] + rm_s[3 * r + 2] * rcw;
    rwn_s[r * 256 + m] = rwn;
    rw[b * 1024 + r * 256 + m] = rwn;
    __syncthreads();
  }

  // read vectors + pack output row
  int rr = tid >> 6, wch = tid & 63;
  float rv = 0.f;
  for (int m2 = 0; m2 < 256; ++m2) rv += rwn_s[rr * 256 + m2] * mem[(size_t)b * 16384 + m2 * 64 + wch];
  float* yrow = ys_t + b * 768;
  yrow[512 + tid] = rv;
  yrow[tid] = clip20(h1[b * 512 + tid]);
  yrow[256 + tid] = clip20(h1[b * 512 + 256 + tid]);
}

// ---- init helpers ----
__global__ void fill_kernel(float* p, float v, int n) {
  int i = blockIdx.x * blockDim.x + threadIdx.x;
  if (i < n) p[i] = v;
}
__global__ void init_state_kernel(const float* __restrict__ h0in, float* h0b, float* c0b,
                                  float* h1b, float* c1b) {
  int i = blockIdx.x * blockDim.x + threadIdx.x;
  if (i < 32768) {
    float a = h0in[i], bb = h0in[32768 + i];
    h0b[i] = a; c0b[i] = a; h1b[i] = bb; c1b[i] = bb;
  }
}

extern "C" void kernel_launch(void* const* d_in, const int* in_sizes, int n_in,
                              void* d_out, int out_size, void* d_ws, size_t ws_size,
                              hipStream_t stream) {
  (void)in_sizes; (void)n_in; (void)out_size; (void)ws_size;
  const float* x    = (const float*)d_in[0];
  const float* h0   = (const float*)d_in[1];
  const float* Wih0 = (const float*)d_in[2];
  const float* Whh0 = (const float*)d_in[3];
  const float* bih0 = (const float*)d_in[4];
  const float* bhh0 = (const float*)d_in[5];
  const float* Wih1 = (const float*)d_in[6];
  const float* Whh1 = (const float*)d_in[7];
  const float* bih1 = (const float*)d_in[8];
  const float* bhh1 = (const float*)d_in[9];
  const float* Wif  = (const float*)d_in[10];
  const float* bif  = (const float*)d_in[11];
  const float* Wout = (const float*)d_in[12];
  const float* bout = (const float*)d_in[13];
  float* out = (float*)d_out;

  float* w = (float*)d_ws;
  float* h0a = w; w += 32768; float* h0b = w; w += 32768;
  float* c0a = w; w += 32768; float* c0b = w; w += 32768;
  float* h1a = w; w += 32768; float* h1b = w; w += 32768;
  float* c1a = w; w += 32768; float* c1b = w; w += 32768;
  float* xi    = w; w += 30720;     // 64 x 480 (cols 471..479 unused)
  float* mem   = w; w += 1048576;   // 64 x 256 x 64
  float* link  = w; w += 4194304;   // 64 x 256 x 256
  float* prec  = w; w += 16384;
  float* rw    = w; w += 65536;     // 64 x 4 x 256
  float* ww    = w; w += 16384;
  float* usage = w; w += 16384;
  float* wwsum = w; w += 64;
  float* hrk   = w; w += 16384;     // 64 x 256
  float* hrs   = w; w += 256;       // 64 x 4
  float* hrm   = w; w += 768;       // 64 x 12
  float* ys    = w; w += 1572864;   // 2048 x 768

  // initial state
  init_state_kernel<<<128, 256, 0, stream>>>(h0, h0a, c0a, h1a, c1a);
  fill_kernel<<<4096, 256, 0, stream>>>(mem, DELTA_, 1048576);
  fill_kernel<<<16384, 256, 0, stream>>>(link, 0.f, 4194304);
  fill_kernel<<<64, 256, 0, stream>>>(prec, 0.f, 16384);
  fill_kernel<<<256, 256, 0, stream>>>(rw, DELTA_, 65536);
  fill_kernel<<<64, 256, 0, stream>>>(ww, DELTA_, 16384);
  fill_kernel<<<64, 256, 0, stream>>>(usage, 0.f, 16384);

  float *h0i = h0a, *h0o = h0b, *c0i = c0a, *c0o = c0b;
  float *h1i = h1a, *h1o = h1b, *c1i = c1a, *c1o = c1b;

  for (int t = 0; t < 32; ++t) {
    // layer 0: inp = [x_t, zeros] -> only first 256 K of Wih0 contribute
    lstm_kernel<<<16, 256, 0, stream>>>(x + t * 256, 8192, 256, Wih0, h0i, Whh0,
                                        bih0, bhh0, c0i, c0o, h0o);
    // layer 1
    lstm_kernel<<<16, 256, 0, stream>>>(h0o, 512, 512, Wih1, h1i, Whh1,
                                        bih1, bhh1, c1i, c1o, h1o);
    // interface vector
    iface_kernel<<<15, 256, 0, stream>>>(h1o, Wif, bif, xi);
    // memory write phase (usage, alloc, ww, erase/write)
    mem_write_kernel<<<64, 256, 0, stream>>>(xi, mem, rw, ww, usage, wwsum, hrk, hrs, hrm);
    // temporal link update (uses OLD prec)
    link_kernel<<<16384, 256, 0, stream>>>(link, ww, prec);
    // precedence update (after link)
    prec_kernel<<<64, 256, 0, stream>>>(prec, ww, wwsum);
    // read phase + pack o_t
    read_kernel<<<64, 256, 0, stream>>>(mem, link, rw, hrk, hrs, hrm, h1o,
                                        ys + (size_t)t * 64 * 768);
    // ping-pong LSTM state
    float* tmp;
    tmp = h0i; h0i = h0o; h0o = tmp;  tmp = c0i; c0i = c0o; c0o = tmp;
    tmp = h1i; h1i = h1o; h1o = tmp;  tmp = c1i; c1i = c1o; c1o = tmp;
  }

  // outputs = tanh(ys) @ Wout^T + bout, laid out (b, T, OUT)
  outgemm_kernel<<<256, 256, 0, stream>>>(ys, Wout, bout, out);
}